// SegmentModel_33277406609649
// MI455X (gfx1250) — compile-verified
//
#include <hip/hip_runtime.h>

#define TB 256      // T
#define BB 16       // B
#define DD 128      // D
#define VV 4096     // V
#define SS 12       // max_seg_len
#define LOGINF 1.0e6f

typedef float v2f __attribute__((ext_vector_type(2)));
typedef float v8f __attribute__((ext_vector_type(8)));

// ---------------------------------------------------------------------------
// Kernel 1: per time-step t (one workgroup per t, 8 waves):
//   start_hidden = tanh(enc @ W_start + b_start)            (WMMA f32 16x16x4)
//   decS = tanh(start_hidden @ W_dec + b_dec)
//   decE = tanh(emb @ W_dec + b_dec)
//   logits = dec @ W_vocab + b_vocab ; online LSE over V ; gather tgt/EOS
// Outputs per (t,b): tlpS = tgt_logit_S - lse_S
//                    tlpE = tgt_logit_E - lse_E
//                    elpE = eos_logit_E - lse_E
// ---------------------------------------------------------------------------
__global__ __launch_bounds__(256) void seg_scores_kernel(
    const float* __restrict__ embeds,      // [B,T,D]
    const float* __restrict__ dec_hidden,  // [B,T,D]
    const float* __restrict__ W_start,     // [D,D]
    const float* __restrict__ b_start,     // [D]
    const float* __restrict__ W_dec,       // [D,D]
    const float* __restrict__ b_dec,       // [D]
    const float* __restrict__ W_vocab,     // [D,V]
    const float* __restrict__ b_vocab,     // [V]
    const int*   __restrict__ x,           // [B,T]
    float* __restrict__ tlpS,
    float* __restrict__ tlpE,
    float* __restrict__ elpE)
{
    const int t    = blockIdx.x;       // 0..T-1
    const int tid  = threadIdx.x;      // 0..255
    const int wave = tid >> 5;         // 0..7
    const int lane = tid & 31;
    const int lhalf = lane >> 4;       // 0/1 (lane half)
    const int llow  = lane & 15;

    __shared__ float sE[BB][DD];       // embedding rows for this t
    __shared__ float sH[BB][DD];       // encoder (decoder_hidden) rows
    __shared__ float sSH[BB][DD];      // start_hidden rows
    __shared__ float sDS[BB][DD];      // decS rows
    __shared__ float sDE[BB][DD];      // decE rows
    __shared__ float sTgt[2][BB];      // target logits (S-rows / E-rows)
    __shared__ float sEos[BB];         // EOS logits (E-rows)
    __shared__ float sMax[2][4][BB];   // per-wave LSE partial max
    __shared__ float sSum[2][4][BB];   // per-wave LSE partial sumexp
    __shared__ float sLz[2][BB];       // final logZ per (mtile,row)
    __shared__ int   sTcol[BB];        // target vocab column per batch row

    // --- stage inputs into LDS ---
    for (int idx = tid; idx < BB * DD; idx += 256) {
        int b = idx >> 7, c = idx & (DD - 1);
        size_t off = ((size_t)b * TB + t) * DD + c;
        sH[b][c] = dec_hidden[off];
        sE[b][c] = embeds[off];
    }
    if (tid < BB) {
        int tn = (t + 1 < TB) ? (t + 1) : (TB - 1);
        sTcol[tid] = x[(size_t)tid * TB + tn];
    }
    __syncthreads();

    // --- Phase A: start_hidden = tanh(enc @ W_start + b_start) ---
    {
        const int n0 = wave * 16;
        v8f acc = {0.f,0.f,0.f,0.f,0.f,0.f,0.f,0.f};
        for (int k0 = 0; k0 < DD; k0 += 4) {
            int ka = k0 + 2 * lhalf;
            v2f a = { sH[llow][ka], sH[llow][ka + 1] };
            v2f bm = { W_start[(size_t)ka * DD + n0 + llow],
                       W_start[(size_t)(ka + 1) * DD + n0 + llow] };
            acc = __builtin_amdgcn_wmma_f32_16x16x4_f32(
                false, a, false, bm, (short)0, acc, false, false);
        }
        const int n = n0 + llow;
        const float bs = b_start[n];
        #pragma unroll
        for (int v = 0; v < 8; ++v) {
            int m = v + 8 * lhalf;
            sSH[m][n] = tanhf(acc[v] + bs);
        }
    }
    __syncthreads();

    // --- Phase B: decS / decE = tanh(row @ W_dec + b_dec) ---
    for (int mt = 0; mt < 2; ++mt) {
        const float (*src)[DD] = (mt == 0) ? sSH : sE;
        float (*dst)[DD]       = (mt == 0) ? sDS : sDE;
        const int n0 = wave * 16;
        v8f acc = {0.f,0.f,0.f,0.f,0.f,0.f,0.f,0.f};
        for (int k0 = 0; k0 < DD; k0 += 4) {
            int ka = k0 + 2 * lhalf;
            v2f a = { src[llow][ka], src[llow][ka + 1] };
            v2f bm = { W_dec[(size_t)ka * DD + n0 + llow],
                       W_dec[(size_t)(ka + 1) * DD + n0 + llow] };
            acc = __builtin_amdgcn_wmma_f32_16x16x4_f32(
                false, a, false, bm, (short)0, acc, false, false);
        }
        const int n = n0 + llow;
        const float bd = b_dec[n];
        #pragma unroll
        for (int v = 0; v < 8; ++v) {
            int m = v + 8 * lhalf;
            dst[m][n] = tanhf(acc[v] + bd);
        }
    }
    __syncthreads();

    // --- Phase C: vocab projection + online logsumexp ---
    const int mw = wave & 1;               // 0 = decS rows, 1 = decE rows
    const int wp = wave >> 1;              // 0..3 (N-tile stride group)
    const float (*Amat)[DD] = mw ? sDE : sDS;

    // A fragments are invariant across all 64 N-tiles: hoist the full K=128
    // fragment set into registers (32 x v2f = 64 VGPRs) so the hot loop has
    // zero LDS traffic — only the two mandatory W_vocab loads per WMMA.
    v2f afrag[DD / 4];
    #pragma unroll
    for (int kk = 0; kk < DD / 4; ++kk) {
        int ka = 4 * kk + 2 * lhalf;
        afrag[kk].x = Amat[llow][ka];
        afrag[kk].y = Amat[llow][ka + 1];
    }

    float rmax[8], rsum[8];
    #pragma unroll
    for (int v = 0; v < 8; ++v) { rmax[v] = -3.0e38f; rsum[v] = 0.f; }

    for (int i = 0; i < 64; ++i) {
        const int nt = wp + 4 * i;         // 0..255
        const int n  = nt * 16 + llow;
        const float* __restrict__ Wp = W_vocab + (size_t)2 * lhalf * VV + n;
        if (i + 1 < 64) {
            // prefetch next tile's B column stream (L2-resident, warm L0)
            __builtin_prefetch(Wp + 64, 0, 3);
        }
        v8f acc = {0.f,0.f,0.f,0.f,0.f,0.f,0.f,0.f};
        #pragma unroll
        for (int kk = 0; kk < DD / 4; ++kk) {
            v2f bm = { Wp[(size_t)(4 * kk) * VV],
                       Wp[(size_t)(4 * kk + 1) * VV] };
            acc = __builtin_amdgcn_wmma_f32_16x16x4_f32(
                false, afrag[kk], false, bm, (short)0, acc, false, false);
        }
        const float bv = b_vocab[n];
        #pragma unroll
        for (int v = 0; v < 8; ++v) {
            float val = acc[v] + bv;
            int m = v + 8 * lhalf;         // row = batch index b
            float nm = fmaxf(rmax[v], val);
            rsum[v] = rsum[v] * __expf(rmax[v] - nm) + __expf(val - nm);
            rmax[v] = nm;
            if (mw == 1 && n == 2) sEos[m] = val;      // EOS_ID == 2
            if (n == sTcol[m])     sTgt[mw][m] = val;  // target token logit
        }
    }

    // reduce LSE partials across the 16 lanes holding each row
    #pragma unroll
    for (int v = 0; v < 8; ++v) {
        float m = rmax[v], s = rsum[v];
        #pragma unroll
        for (int off = 1; off < 16; off <<= 1) {
            float mo = __shfl_xor(m, off, 32);
            float so = __shfl_xor(s, off, 32);
            float nm = fmaxf(m, mo);
            s = s * __expf(m - nm) + so * __expf(mo - nm);
            m = nm;
        }
        if (llow == 0) {
            sMax[mw][wp][v + 8 * lhalf] = m;
            sSum[mw][wp][v + 8 * lhalf] = s;
        }
    }
    __syncthreads();

    // combine the 4 wave partials per (mtile,row)
    if (tid < 32) {
        int mm = tid >> 4, r = tid & 15;
        float m = sMax[mm][0][r], s = sSum[mm][0][r];
        #pragma unroll
        for (int p = 1; p < 4; ++p) {
            float mo = sMax[mm][p][r], so = sSum[mm][p][r];
            float nm = fmaxf(m, mo);
            s = s * __expf(m - nm) + so * __expf(mo - nm);
            m = nm;
        }
        sLz[mm][r] = m + __logf(s);
    }
    __syncthreads();

    if (tid < BB) {
        int b = tid;
        int o = t * BB + b;
        tlpS[o] = sTgt[0][b] - sLz[0][b];
        tlpE[o] = sTgt[1][b] - sLz[1][b];
        elpE[o] = sEos[b]    - sLz[1][b];
    }
}

// ---------------------------------------------------------------------------
// Kernel 2: alpha recursion DP + final reduction (tiny, sequential in T).
// logpy[j,l,b] = tlpS[j-1] + sum_{u=j..j+l-1} tlpE[u]
//             + (l>=1 ? sum_{u=j..j+l} pen[u] : 0) + elpE[j+l]
// alpha[je] = LSE_l( alpha[je-1-l] + logpy[je-l, l] ),  eos term factored out.
// ---------------------------------------------------------------------------
__global__ __launch_bounds__(32) void seg_dp_kernel(
    const int*   __restrict__ x,        // [B,T]
    const int*   __restrict__ lengths,  // [B]
    const float* __restrict__ tlpS,     // [T,B]
    const float* __restrict__ tlpE,     // [T,B]
    const float* __restrict__ elpE,     // [T,B]
    float* __restrict__ out)
{
    __shared__ float alpha[BB][TB];
    __shared__ float cumE[BB][TB];
    __shared__ float cumP[BB][TB];
    __shared__ float nllBuf[BB];
    __shared__ float lenBuf[BB];

    const int b = threadIdx.x;
    if (b < BB) {
        // prefix sums: token log-probs (E rows) and is_single penalty
        float ce = 0.f, cp = 0.f;
        for (int u = 0; u < TB; ++u) {
            ce += tlpE[u * BB + b];
            int tok = x[(size_t)b * TB + u];
            if (tok == 3 || tok == 4 || tok == 5) cp -= LOGINF;  // PUNC/ENG/NUM
            cumE[b][u] = ce;
            cumP[b][u] = cp;
        }
        alpha[b][0] = 0.f;
        for (int je = 1; je <= TB - 2; ++je) {
            float vals[SS];
            const int L = (SS < je) ? SS : je;   // l = 0..L-1, jstart >= 0
            float m = -3.0e38f;
            for (int l = 0; l < L; ++l) {
                int j = je - l;                  // segment first-token index
                float v = alpha[b][j - 1]
                        + tlpS[(j - 1) * BB + b]
                        + (cumE[b][je - 1] - cumE[b][j - 1])
                        + ((l >= 1) ? (cumP[b][je] - cumP[b][je - l - 1]) : 0.f);
                vals[l] = v;
                m = fmaxf(m, v);
            }
            float s = 0.f;
            for (int l = 0; l < L; ++l) s += __expf(vals[l] - m);
            alpha[b][je] = m + __logf(s) + elpE[je * BB + b];
        }
        int len = lengths[b];
        nllBuf[b] = fmaxf(-alpha[b][len - 2], 0.f);
        lenBuf[b] = (float)(len - 2);
    }
    __syncthreads();
    if (b == 0) {
        float sn = 0.f, st = 0.f;
        for (int i = 0; i < BB; ++i) { sn += nllBuf[i]; st += lenBuf[i]; }
        out[0] = sn / st;
    }
}

// ---------------------------------------------------------------------------
extern "C" void kernel_launch(void* const* d_in, const int* in_sizes, int n_in,
                              void* d_out, int out_size, void* d_ws, size_t ws_size,
                              hipStream_t stream) {
    const float* embeds     = (const float*)d_in[0];
    const float* dec_hidden = (const float*)d_in[1];
    const float* W_start    = (const float*)d_in[2];
    const float* b_start    = (const float*)d_in[3];
    const float* W_dec      = (const float*)d_in[4];
    const float* b_dec      = (const float*)d_in[5];
    const float* W_vocab    = (const float*)d_in[6];
    const float* b_vocab    = (const float*)d_in[7];
    const int*   x          = (const int*)d_in[8];
    const int*   lengths    = (const int*)d_in[9];
    (void)in_sizes; (void)n_in; (void)out_size; (void)ws_size;

    float* ws   = (float*)d_ws;
    float* tlpS = ws;                 // [T,B]
    float* tlpE = ws + TB * BB;       // [T,B]
    float* elpE = ws + 2 * TB * BB;   // [T,B]

    seg_scores_kernel<<<TB, 256, 0, stream>>>(
        embeds, dec_hidden, W_start, b_start, W_dec, b_dec,
        W_vocab, b_vocab, x, tlpS, tlpE, elpE);

    seg_dp_kernel<<<1, 32, 0, stream>>>(
        x, lengths, tlpS, tlpE, elpE, (float*)d_out);
}